// LINK_55688545960300
// MI455X (gfx1250) — compile-verified
//
#include <hip/hip_runtime.h>

typedef __attribute__((ext_vector_type(2))) float v2f;
typedef __attribute__((ext_vector_type(8))) float v8f;

// ---------------------------------------------------------------------------
// Kernel 1: initialize out[n,c] = b[c] on the matrix pipe.
// D = A(16x4, all ones) x B(4x16, every row = b/4) + 0  ==> D[m,n] = b[n].
// Exact in f32 (1.0 * b, RNE). One wave handles 16 nodes x 64 channels
// (4 WMMA tiles). Full tiles take an unguarded store path (wave-uniform
// branch) so only the single boundary wave pays exec-mask juggling.
// ---------------------------------------------------------------------------
__global__ void bias_init_wmma(const float* __restrict__ bias,
                               float* __restrict__ out,
                               int N, int* __restrict__ rowmin) {
  if (blockIdx.x == 0 && threadIdx.x == 0) *rowmin = 0x7fffffff;

  const int wave = threadIdx.x >> 5;          // 8 waves / block
  const int lane = threadIdx.x & 31;
  const int nodebase = (blockIdx.x * 8 + wave) * 16;
  if (nodebase >= N) return;                  // wave-uniform

  const int half = lane >> 4;                 // 0: M 0-7, 1: M 8-15
  const int n16  = lane & 15;                 // N within tile
  const bool full = (nodebase + 16) <= N;     // wave-uniform

  v2f a; a.x = 1.0f; a.y = 1.0f;              // A = all ones (layout-agnostic)

#pragma unroll
  for (int t = 0; t < 4; ++t) {
    const int colbase = t * 16;
    const float bv = bias[colbase + n16] * 0.25f;  // 4 K-rows contribute b/4 each
    v2f bb; bb.x = bv; bb.y = bv;
    v8f c = {};
    v8f d = __builtin_amdgcn_wmma_f32_16x16x4_f32(
        /*neg_a=*/false, a, /*neg_b=*/false, bb,
        /*c_mod=*/(short)0, c, /*reuse_a=*/false, /*reuse_b=*/false);

    float* base = out + (size_t)(nodebase + half * 8) * 64 + colbase + n16;
    if (full) {
#pragma unroll
      for (int j = 0; j < 8; ++j) base[(size_t)j * 64] = d[j];
    } else {
#pragma unroll
      for (int j = 0; j < 8; ++j) {
        if (nodebase + half * 8 + j < N) base[(size_t)j * 64] = d[j];
      }
    }
  }
}

// ---------------------------------------------------------------------------
// Kernel 2: row minimum (reference does row - row.min()).
// ---------------------------------------------------------------------------
__global__ void row_min_kernel(const long long* __restrict__ rows, int E,
                               int* __restrict__ rowmin) {
  int v = 0x7fffffff;
  for (long long e = (long long)blockIdx.x * blockDim.x + threadIdx.x; e < E;
       e += (long long)gridDim.x * blockDim.x) {
    const int r = (int)rows[e];
    v = r < v ? r : v;
  }
  atomicMin(rowmin, v);
}

// ---------------------------------------------------------------------------
// Kernel 3: transpose W[64,N] -> Wt[N,64] through a padded LDS tile; both
// global phases fully coalesced, LDS reads bank-conflict free (stride 65).
// ---------------------------------------------------------------------------
__global__ void transpose_w(const float* __restrict__ W,
                            float* __restrict__ Wt, int N) {
  __shared__ float tile[64][65];
  const int cbase = blockIdx.x * 64;

#pragma unroll
  for (int i = 0; i < 16; ++i) {
    const int lin = i * 256 + threadIdx.x;
    const int r = lin >> 6, c = lin & 63;
    const int gc = cbase + c;
    tile[r][c] = (gc < N) ? W[(size_t)r * N + gc] : 0.0f;
  }
  __syncthreads();
#pragma unroll
  for (int i = 0; i < 16; ++i) {
    const int lin = i * 256 + threadIdx.x;
    const int nl = lin >> 6, ch = lin & 63;
    const int gn = cbase + nl;
    if (gn < N) Wt[(size_t)gn * 64 + ch] = tile[ch][nl];
  }
}

// ---------------------------------------------------------------------------
// Kernel 4: edge scatter, batched 32 edges per wave32.
//  - lanes make ONE coalesced load of 32 edge (row,col) pairs,
//  - each edge is broadcast wave-wide via __shfl,
//  - all 32 lanes gather the contiguous 256 B Wt row (float2/lane, L2-hit)
//    and issue two global_atomic_add_f32 into the destination row.
// Prefetch the next batch's indices behind the atomic stream.
// ---------------------------------------------------------------------------
__global__ void scatter_edges(const long long* __restrict__ rows,
                              const long long* __restrict__ cols,
                              const float* __restrict__ Wt,
                              float* __restrict__ out,
                              const int* __restrict__ rowmin, int E) {
  const int lane = threadIdx.x & 31;
  const int wave = blockIdx.x * (blockDim.x >> 5) + (threadIdx.x >> 5);
  const int nwaves = gridDim.x * (blockDim.x >> 5);
  const int rmin = *rowmin;
  const int stride = nwaves * 32;

  for (int base = wave * 32; base < E; base += stride) {
    const int eidx = base + lane;
    int rv = 0, cv = 0;
    if (eidx < E) {
      rv = (int)rows[eidx] - rmin;
      cv = (int)cols[eidx];
    }
    if (base + stride < E) {  // speculative prefetch of next batch
      __builtin_prefetch(&rows[base + stride + lane], 0, 0);
      __builtin_prefetch(&cols[base + stride + lane], 0, 0);
    }
    const int cnt = (E - base) < 32 ? (E - base) : 32;
    for (int k = 0; k < cnt; ++k) {
      const int r = __shfl(rv, k, 32);
      const int c = __shfl(cv, k, 32);
      const float2 w = ((const float2*)(Wt + (size_t)c * 64))[lane];
      float* dst = out + (size_t)r * 64 + lane * 2;
      atomicAdd(dst, w.x);
      atomicAdd(dst + 1, w.y);
    }
  }
}

// Fallback when workspace cannot hold Wt: gather W columns with stride N.
__global__ void scatter_edges_strided(const long long* __restrict__ rows,
                                      const long long* __restrict__ cols,
                                      const float* __restrict__ W,
                                      float* __restrict__ out,
                                      const int* __restrict__ rowmin,
                                      int E, int N) {
  const int lane = threadIdx.x & 31;
  const int wave = blockIdx.x * (blockDim.x >> 5) + (threadIdx.x >> 5);
  const int nwaves = gridDim.x * (blockDim.x >> 5);
  const int rmin = *rowmin;

  for (int e = wave; e < E; e += nwaves) {
    const int r = (int)rows[e] - rmin;
    const int c = (int)cols[e];
    const float w0 = W[(size_t)(2 * lane) * N + c];
    const float w1 = W[(size_t)(2 * lane + 1) * N + c];
    float* dst = out + (size_t)r * 64 + lane * 2;
    atomicAdd(dst, w0);
    atomicAdd(dst + 1, w1);
  }
}

extern "C" void kernel_launch(void* const* d_in, const int* in_sizes, int n_in,
                              void* d_out, int out_size, void* d_ws, size_t ws_size,
                              hipStream_t stream) {
  const long long* edges = (const long long*)d_in[0];  // [2, E] int64
  const float* W = (const float*)d_in[1];              // [64, N] f32
  const float* bias = (const float*)d_in[2];           // [64] f32
  float* out = (float*)d_out;                          // [N, 64] f32

  const int E = in_sizes[0] / 2;
  const int C = in_sizes[2];                           // 64
  const int N = in_sizes[1] / C;

  int* rowmin = (int*)d_ws;
  float* Wt = (float*)((char*)d_ws + 256);
  const size_t need = 256 + (size_t)N * 64 * sizeof(float);

  const long long* rows = edges;
  const long long* cols = edges + E;

  // 1) out = broadcast(b) via WMMA; seeds rowmin scratch.
  bias_init_wmma<<<(N + 127) / 128, 256, 0, stream>>>(bias, out, N, rowmin);
  // 2) rowmin = min(row)
  row_min_kernel<<<256, 256, 0, stream>>>(rows, E, rowmin);

  if (ws_size >= need) {
    // 3) Wt = W^T (coalesced LDS tile transpose)
    transpose_w<<<(N + 63) / 64, 256, 0, stream>>>(W, Wt, N);
    // 4) scatter-add messages (L2-atomic bound)
    scatter_edges<<<2048, 256, 0, stream>>>(rows, cols, Wt, out, rowmin, E);
  } else {
    scatter_edges_strided<<<2048, 256, 0, stream>>>(rows, cols, W, out, rowmin, E, N);
  }
}